// VariationalGCNEncoder_5686536700334
// MI455X (gfx1250) — compile-verified
//
#include <hip/hip_runtime.h>
#include <hip/hip_bf16.h>
#include <stdint.h>

typedef __attribute__((ext_vector_type(2))) float v2f;
typedef __attribute__((ext_vector_type(8))) float v8f;

#define N_NODES 50000
#define IN_CH   512
#define HID_CH  128
#define OUT_CH  64
#define ROWTILES (N_NODES / 16)   // 3125, exact
#define PSTRIDE1 288              // words per pair-row, == 32 mod 64 -> conflict-free b64 reads
#define PSTRIDE2 160              // words per pair-row, == 32 mod 64

static __device__ __forceinline__ void atomAddF(float* p, float v) {
    (void)__hip_atomic_fetch_add(p, v, __ATOMIC_RELAXED, __HIP_MEMORY_SCOPE_AGENT);
}

// Async global->LDS copies (CDNA5): per-lane LDS dest address, tracked by ASYNCcnt.
static __device__ __forceinline__ void async_g2lds_b32(unsigned lds_byte_addr, const float* g) {
    asm volatile("global_load_async_to_lds_b32 %0, %1, off"
                 :: "v"(lds_byte_addr), "v"((unsigned long long)(uintptr_t)g)
                 : "memory");
}
static __device__ __forceinline__ void wait_async0() {
    asm volatile("s_wait_asynccnt 0x0" ::: "memory");
}

// ---------------- degree / normalization ----------------
__global__ void k_init_deg(float* __restrict__ deg) {
    int i = blockIdx.x * blockDim.x + threadIdx.x;
    if (i < N_NODES) deg[i] = 1.0f;              // self-loop contributes 1
}

__global__ void k_scatter_deg(const int* __restrict__ ei, int nE, float* __restrict__ deg) {
    int e = blockIdx.x * blockDim.x + threadIdx.x;
    if (e < nE) atomAddF(&deg[ei[nE + e]], 1.0f); // dst row of edge_index
}

__global__ void k_dinv(float* __restrict__ deg) {
    int i = blockIdx.x * blockDim.x + threadIdx.x;
    if (i < N_NODES) deg[i] = rsqrtf(deg[i]);     // deg >= 1 always
}

// ---------------- GEMM1: xw = x @ W1 (fp32 WMMA 16x16x4) ----------------
// 8 waves/block, one 16-row tile each. W1 staged pair-packed into LDS via async b32
// scatter (two 256-row phases, 147 KB): element (row 2q+j, col c) -> word q*288+2c+j.
// Each WMMA B fragment is then ONE aligned ds_load_b64 (no packing moves).
__global__ void __launch_bounds__(256) k_gemm1(const float* __restrict__ X,
                                               const float* __restrict__ W,
                                               float* __restrict__ XW) {
    __shared__ float Bs[128 * PSTRIDE1];          // 147,456 B
    const int tid  = threadIdx.x;
    const int wave = tid >> 5;
    const int lane = tid & 31;
    const int m = lane & 15, half = lane >> 4;

    const int myTile = blockIdx.x * 8 + wave;
    const bool active = (myTile < ROWTILES);      // wave-uniform: EXEC stays all-ones
    const int row0 = myTile * 16;
    const float* xr = X + (size_t)((active ? row0 : 0) + m) * IN_CH + 2 * half;

    // staging coords: thread = (column sc 0..127, row-parity sj 0..1)
    const int sc = tid & 127;
    const int sj = tid >> 7;
    v8f acc[8] = {};

    for (int ph = 0; ph < 2; ++ph) {
        __syncthreads();                          // previous-phase LDS reads done
        const float* gbase = W + (size_t)(ph * 256 + sj) * HID_CH + sc;
        for (int i = 0; i < 128; ++i) {           // pair-rows q = 0..127
            unsigned lb = (unsigned)(uintptr_t)(&Bs[i * PSTRIDE1 + 2 * sc + sj]);
            async_g2lds_b32(lb, gbase + (size_t)i * 2 * HID_CH);
        }
        wait_async0();
        __syncthreads();

        if (active) {
            const float* xp = xr + ph * 256;
            for (int k = 0; k < 256; k += 4) {
                v2f a = __builtin_nontemporal_load((const v2f*)(xp + k)); // x streamed once
                int base = ((k >> 1) + half) * PSTRIDE1 + 2 * m;          // pair q = k/2 + half
#pragma unroll
                for (int t = 0; t < 8; ++t) {
                    v2f b = *(const v2f*)(&Bs[base + 32 * t]);            // one ds_load_b64
                    acc[t] = __builtin_amdgcn_wmma_f32_16x16x4_f32(
                        false, a, false, b, (short)0, acc[t], false, false);
                }
            }
        }
    }

    if (active) {
#pragma unroll
        for (int t = 0; t < 8; ++t) {
            float* orow = XW + (size_t)(row0 + 8 * half) * HID_CH + t * 16 + m;
#pragma unroll
            for (int r = 0; r < 8; ++r)           // C layout: VGPR r -> rows r / r+8
                orow[(size_t)r * HID_CH] = acc[t][r];
        }
    }
}

// ---------------- h init: bias + self-loop term ----------------
__global__ void k_init_h(const float* __restrict__ XW, const float* __restrict__ dinv,
                         const float* __restrict__ b1, float* __restrict__ H) {
    int idx = blockIdx.x * blockDim.x + threadIdx.x;    // over N*128/4
    if (idx >= N_NODES * (HID_CH / 4)) return;
    int i  = idx / (HID_CH / 4);
    int c4 = (idx % (HID_CH / 4)) * 4;
    float d  = dinv[i];
    float d2 = d * d;
    float4 xv = *(const float4*)(XW + (size_t)i * HID_CH + c4);
    float4 bv = *(const float4*)(b1 + c4);
    float4 o;
    o.x = bv.x + d2 * xv.x; o.y = bv.y + d2 * xv.y;
    o.z = bv.z + d2 * xv.z; o.w = bv.w + d2 * xv.w;
    *(float4*)(H + (size_t)i * HID_CH + c4) = o;
}

// ---------------- conv1 edge scatter: wave per edge, 128 channels ----------------
__global__ void k_scatter_h(const int* __restrict__ ei, int nE, const float* __restrict__ dinv,
                            const float* __restrict__ XW, float* __restrict__ H) {
    int e    = (blockIdx.x * blockDim.x + threadIdx.x) >> 5;
    int lane = threadIdx.x & 31;
    if (e >= nE) return;
    int s = ei[e], d = ei[nE + e];
    float nm = dinv[s] * dinv[d];
    float4 xv = *(const float4*)(XW + (size_t)s * HID_CH + lane * 4);  // coalesced 512B gather
    float* hp = H + (size_t)d * HID_CH + lane * 4;
    atomAddF(hp + 0, nm * xv.x);
    atomAddF(hp + 1, nm * xv.y);
    atomAddF(hp + 2, nm * xv.z);
    atomAddF(hp + 3, nm * xv.w);
}

__global__ void k_relu(float* __restrict__ H, int n4) {
    int idx = blockIdx.x * blockDim.x + threadIdx.x;
    if (idx >= n4) return;
    float4 v = *(float4*)(H + (size_t)idx * 4);
    v.x = fmaxf(v.x, 0.f); v.y = fmaxf(v.y, 0.f);
    v.z = fmaxf(v.z, 0.f); v.w = fmaxf(v.w, 0.f);
    *(float4*)(H + (size_t)idx * 4) = v;
}

// ---------------- GEMM2: hw_mu = h@W_mu, hw_ls = h@W_logstd (pair-packed LDS B) ----------------
__global__ void __launch_bounds__(256) k_gemm2(const float* __restrict__ H,
                                               const float* __restrict__ Wmu,
                                               const float* __restrict__ Wls,
                                               float* __restrict__ HMu,
                                               float* __restrict__ HLs) {
    __shared__ float Bs[2 * 64 * PSTRIDE2];       // 81,920 B: [0]=mu, [64*160]=logstd
    const int tid  = threadIdx.x;
    const int wave = tid >> 5;
    const int lane = tid & 31;
    const int m = lane & 15, half = lane >> 4;

    const int myTile = blockIdx.x * 8 + wave;
    const bool active = (myTile < ROWTILES);
    const int row0 = myTile * 16;

    // stage pair-packed: thread = (col c 0..63, parity j 0..1, q offset 0..1)
    {
        int c  = tid & 63;
        int j  = (tid >> 6) & 1;
        int qo = tid >> 7;
        for (int i = 0; i < 32; ++i) {
            int q = i * 2 + qo;                   // pair-row 0..63
            unsigned lbm = (unsigned)(uintptr_t)(&Bs[q * PSTRIDE2 + 2 * c + j]);
            unsigned lbl = (unsigned)(uintptr_t)(&Bs[64 * PSTRIDE2 + q * PSTRIDE2 + 2 * c + j]);
            async_g2lds_b32(lbm, Wmu + ((size_t)(2 * q + j) * OUT_CH + c));
            async_g2lds_b32(lbl, Wls + ((size_t)(2 * q + j) * OUT_CH + c));
        }
        wait_async0();
        __syncthreads();
    }

    if (!active) return;
    const float* hr = H + (size_t)(row0 + m) * HID_CH + 2 * half;
    v8f amu[4] = {};
    v8f als[4] = {};

    for (int k = 0; k < HID_CH; k += 4) {
        v2f a = *(const v2f*)(hr + k);
        int base = ((k >> 1) + half) * PSTRIDE2 + 2 * m;
#pragma unroll
        for (int t = 0; t < 4; ++t) {
            v2f b = *(const v2f*)(&Bs[base + 32 * t]);
            amu[t] = __builtin_amdgcn_wmma_f32_16x16x4_f32(
                false, a, false, b, (short)0, amu[t], false, false);
            v2f c2 = *(const v2f*)(&Bs[64 * PSTRIDE2 + base + 32 * t]);
            als[t] = __builtin_amdgcn_wmma_f32_16x16x4_f32(
                false, a, false, c2, (short)0, als[t], false, false);
        }
    }
#pragma unroll
    for (int t = 0; t < 4; ++t) {
        float* om = HMu + (size_t)(row0 + 8 * half) * OUT_CH + t * 16 + m;
        float* ol = HLs + (size_t)(row0 + 8 * half) * OUT_CH + t * 16 + m;
#pragma unroll
        for (int r = 0; r < 8; ++r) {
            om[(size_t)r * OUT_CH] = amu[t][r];
            ol[(size_t)r * OUT_CH] = als[t][r];
        }
    }
}

// ---------------- out init: bias + self-loop for mu & logstd ----------------
__global__ void k_init_out(const float* __restrict__ HMu, const float* __restrict__ HLs,
                           const float* __restrict__ dinv, const float* __restrict__ bmu,
                           const float* __restrict__ bls, float* __restrict__ out) {
    int idx = blockIdx.x * blockDim.x + threadIdx.x;   // over N*64
    if (idx >= N_NODES * OUT_CH) return;
    int i = idx / OUT_CH, c = idx % OUT_CH;
    float d  = dinv[i];
    float d2 = d * d;
    out[idx] = bmu[c] + d2 * HMu[idx];
    out[(size_t)N_NODES * OUT_CH + idx] = bls[c] + d2 * HLs[idx];
}

// ---------------- conv2 edge scatter: wave per edge, 64+64 channels ----------------
__global__ void k_scatter_out(const int* __restrict__ ei, int nE, const float* __restrict__ dinv,
                              const float* __restrict__ HMu, const float* __restrict__ HLs,
                              float* __restrict__ out) {
    int e    = (blockIdx.x * blockDim.x + threadIdx.x) >> 5;
    int lane = threadIdx.x & 31;
    if (e >= nE) return;
    int s = ei[e], d = ei[nE + e];
    float nm = dinv[s] * dinv[d];
    float2 mv = *(const float2*)(HMu + (size_t)s * OUT_CH + lane * 2);
    float2 lv = *(const float2*)(HLs + (size_t)s * OUT_CH + lane * 2);
    float* mp = out + (size_t)d * OUT_CH + lane * 2;
    float* lp = out + (size_t)N_NODES * OUT_CH + (size_t)d * OUT_CH + lane * 2;
    atomAddF(mp + 0, nm * mv.x);
    atomAddF(mp + 1, nm * mv.y);
    atomAddF(lp + 0, nm * lv.x);
    atomAddF(lp + 1, nm * lv.y);
}

extern "C" void kernel_launch(void* const* d_in, const int* in_sizes, int n_in,
                              void* d_out, int out_size, void* d_ws, size_t ws_size,
                              hipStream_t stream) {
    const float* x   = (const float*)d_in[0];
    const int*   ei  = (const int*)d_in[1];     // [2, E] flattened: src then dst
    const float* W1  = (const float*)d_in[2];
    const float* b1  = (const float*)d_in[3];
    const float* Wmu = (const float*)d_in[4];
    const float* bmu = (const float*)d_in[5];
    const float* Wls = (const float*)d_in[6];
    const float* bls = (const float*)d_in[7];
    float*       out = (float*)d_out;

    const int E = in_sizes[1] / 2;

    // Workspace layout (floats). xw region is reused for hw_mu/hw_logstd after conv1.
    float* ws   = (float*)d_ws;
    float* dinv = ws;                            // 50000
    float* xw   = ws + 50176;                    // 6,400,000  (N*128)
    float* h    = ws + 50176 + 6400000;          // 6,400,000  (N*128)
    float* hwmu = xw;                            // 3,200,000  (N*64)  reuse
    float* hwls = xw + 3200000;                  // 3,200,000  (N*64)  reuse

    const int B = 256;
    const int gemmBlocks = (ROWTILES + 7) / 8;   // 391

    // 1) degrees + rsqrt
    k_init_deg   <<<(N_NODES + B - 1) / B, B, 0, stream>>>(dinv);
    k_scatter_deg<<<(E + B - 1) / B,       B, 0, stream>>>(ei, E, dinv);
    k_dinv       <<<(N_NODES + B - 1) / B, B, 0, stream>>>(dinv);

    // 2) xw = x @ W1  (WMMA, async pair-packed LDS B)
    k_gemm1<<<gemmBlocks, B, 0, stream>>>(x, W1, xw);

    // 3) conv1 aggregate: init (bias + self loop) -> edge scatter -> relu
    k_init_h   <<<(N_NODES * (HID_CH / 4) + B - 1) / B, B, 0, stream>>>(xw, dinv, b1, h);
    k_scatter_h<<<(E + 7) / 8,                          B, 0, stream>>>(ei, E, dinv, xw, h);
    k_relu     <<<(N_NODES * (HID_CH / 4) + B - 1) / B, B, 0, stream>>>(h, N_NODES * (HID_CH / 4));

    // 4) hw_mu = h @ W_mu ; hw_ls = h @ W_logstd  (WMMA, shared A, pair-packed LDS B)
    k_gemm2<<<gemmBlocks, B, 0, stream>>>(h, Wmu, Wls, hwmu, hwls);

    // 5) conv2 aggregate straight into d_out
    k_init_out   <<<(N_NODES * OUT_CH + B - 1) / B, B, 0, stream>>>(hwmu, hwls, dinv, bmu, bls, out);
    k_scatter_out<<<(E + 7) / 8,                    B, 0, stream>>>(ei, E, dinv, hwmu, hwls, out);
}